// Network_47888885350936
// MI455X (gfx1250) — compile-verified
//
#include <hip/hip_runtime.h>
#include <stdint.h>

// Problem constants (from the reference): D = 256, N = 4096.
#define DDIM 256
#define KB   16                 // k-rows per block
#define TPB  256                // one thread per output column j (8 wave32s)

#define AS1 __attribute__((address_space(1)))
#define AS3 __attribute__((address_space(3)))

typedef int v4i __attribute__((ext_vector_type(4)));

// ---------------------------------------------------------------------------
// Async global->LDS staging (CDNA5 path, tracked with ASYNCcnt).
// Each of the 256 threads copies 16B per step; one step moves 4 KB.
// Builtin signature (confirmed by hipcc diagnostics): (v4i AS1* src,
// v4i AS3* dst, imm offset, imm cpol).
// ---------------------------------------------------------------------------
#if defined(__AMDGCN__) && __has_builtin(__builtin_amdgcn_global_load_async_to_lds_b128)
#define HAVE_ASYNC_LDS 1
template <int IT, int NIT>
struct AsyncTile {
    static __device__ inline void run(AS1 v4i* g, AS3 v4i* l) {
        // INST_OFFSET is added to both the global and the LDS address (ISA 08 §4.4)
        __builtin_amdgcn_global_load_async_to_lds_b128(g, l, IT * (TPB * 16), 0);
        AsyncTile<IT + 1, NIT>::run(g, l);
    }
};
template <int NIT>
struct AsyncTile<NIT, NIT> {
    static __device__ inline void run(AS1 v4i*, AS3 v4i*) {}
};
#else
#define HAVE_ASYNC_LDS 0
#endif

// ---------------------------------------------------------------------------
// Main kernel: out[k,j] = max( max_i( W[i,j]*x0[k,i] + BC[i,j] ), x1[k,j] )
// Thread t owns column j = t for KB consecutive k-rows.
//   - x0 tile (KB x 256 f32, contiguous 16 KB) staged via async-to-LDS.
//   - x0 values read back as ds_load_b128 broadcasts (uniform address).
//   - W/BC column loads are coalesced across j (consecutive addresses).
//   - i unrolled by 4 (+pragma x2); max folding written to form v_max3.
// If HAS_C: BCp is B and Cp is C_norm (no workspace); else BCp is precomputed.
// ---------------------------------------------------------------------------
template <bool HAS_C>
__global__ __launch_bounds__(TPB) void maxfma_kernel(const float* __restrict__ x0,
                                                     const float* __restrict__ x1,
                                                     const float* __restrict__ W,
                                                     const float* __restrict__ BCp,
                                                     const float* __restrict__ Cp,
                                                     float* __restrict__ out) {
    __shared__ float x0t[KB * DDIM];          // 16 KB tile of x0 rows k0..k0+KB
    const int j  = threadIdx.x;
    const int k0 = blockIdx.x * KB;

    // ---- Stage x0[k0 : k0+KB, 0:256] (flat contiguous 16 KB) into LDS ----
    {
        const float* gsrc = x0 + (size_t)k0 * DDIM;
#if HAVE_ASYNC_LDS
        AS1 v4i* g = (AS1 v4i*)((char*)const_cast<float*>(gsrc) + j * 16);
        AS3 v4i* l = (AS3 v4i*)((char*)x0t + j * 16);
        AsyncTile<0, (KB * DDIM * 4) / (TPB * 16)>::run(g, l);   // 4 async b128 ops/thread
#if __has_builtin(__builtin_amdgcn_s_wait_asynccnt)
        __builtin_amdgcn_s_wait_asynccnt(0);
#else
        asm volatile("s_wait_asynccnt 0" ::: "memory");
#endif
#else
        float4*       dst = (float4*)x0t;
        const float4* src = (const float4*)gsrc;
        for (int it = j; it < KB * DDIM / 4; it += TPB) dst[it] = src[it];
#endif
        __syncthreads();
    }

    // ---- Init accumulators with the x1 slab (it participates in the max) ----
    float acc[KB];
#pragma unroll
    for (int kk = 0; kk < KB; ++kk) acc[kk] = x1[(size_t)(k0 + kk) * DDIM + j];

    // ---- Main reduction over i, unrolled by 4 in code, x2 by pragma ----
#pragma unroll 2
    for (int i4 = 0; i4 < DDIM / 4; ++i4) {
        const int i = i4 * 4;
        const float w0 = W[(i + 0) * DDIM + j];
        const float w1 = W[(i + 1) * DDIM + j];
        const float w2 = W[(i + 2) * DDIM + j];
        const float w3 = W[(i + 3) * DDIM + j];
        float b0 = BCp[(i + 0) * DDIM + j];
        float b1 = BCp[(i + 1) * DDIM + j];
        float b2 = BCp[(i + 2) * DDIM + j];
        float b3 = BCp[(i + 3) * DDIM + j];
        if (HAS_C) {                                // no workspace: fold C here
            b0 += Cp[(i + 0) * DDIM + j];
            b1 += Cp[(i + 1) * DDIM + j];
            b2 += Cp[(i + 2) * DDIM + j];
            b3 += Cp[(i + 3) * DDIM + j];
        }
#pragma unroll
        for (int kk = 0; kk < KB; ++kk) {
            const float4 xv = *(const float4*)&x0t[kk * DDIM + i];  // ds_load_b128 broadcast
            const float t0 = fmaf(w0, xv.x, b0);
            const float t1 = fmaf(w1, xv.y, b1);
            const float t2 = fmaf(w2, xv.z, b2);
            const float t3 = fmaf(w3, xv.w, b3);
            acc[kk] = fmaxf(fmaxf(acc[kk], t0), t1);   // -> v_max3
            acc[kk] = fmaxf(fmaxf(acc[kk], t2), t3);   // -> v_max3
        }
    }

    // ---- Coalesced store ----
#pragma unroll
    for (int kk = 0; kk < KB; ++kk) out[(size_t)(k0 + kk) * DDIM + j] = acc[kk];
}

// Force the hot kernel to be emitted first so the disasm snippet shows it.
template __global__ void maxfma_kernel<false>(const float* __restrict__,
                                              const float* __restrict__,
                                              const float* __restrict__,
                                              const float* __restrict__,
                                              const float* __restrict__,
                                              float* __restrict__);
template __global__ void maxfma_kernel<true>(const float* __restrict__,
                                             const float* __restrict__,
                                             const float* __restrict__,
                                             const float* __restrict__,
                                             const float* __restrict__,
                                             float* __restrict__);

// ---------------------------------------------------------------------------
// BC = B + C_norm (256x256, trivially memory bound, runs once per launch
// into the workspace).
// ---------------------------------------------------------------------------
__global__ __launch_bounds__(TPB) void bc_precompute(const float* __restrict__ B,
                                                     const float* __restrict__ C,
                                                     float* __restrict__ BC) {
    const int idx = blockIdx.x * TPB + threadIdx.x;
    BC[idx] = B[idx] + C[idx];
}

// ---------------------------------------------------------------------------
// Host entry. Input order from setup_inputs(): x0, x1, C_norm, W, B.
// ---------------------------------------------------------------------------
extern "C" void kernel_launch(void* const* d_in, const int* in_sizes, int n_in,
                              void* d_out, int out_size, void* d_ws, size_t ws_size,
                              hipStream_t stream) {
    const float* x0 = (const float*)d_in[0];
    const float* x1 = (const float*)d_in[1];
    const float* C  = (const float*)d_in[2];
    const float* W  = (const float*)d_in[3];
    const float* B  = (const float*)d_in[4];
    float*       out = (float*)d_out;

    const int n = in_sizes[0] / DDIM;      // 4096 rows
    const int nblk = n / KB;               // 256 blocks

    const size_t bc_bytes = (size_t)DDIM * DDIM * sizeof(float);
    if (ws_size >= bc_bytes) {
        float* BC = (float*)d_ws;
        bc_precompute<<<(DDIM * DDIM) / TPB, TPB, 0, stream>>>(B, C, BC);
        maxfma_kernel<false><<<nblk, TPB, 0, stream>>>(x0, x1, W, BC, nullptr, out);
    } else {
        maxfma_kernel<true><<<nblk, TPB, 0, stream>>>(x0, x1, W, B, C, out);
    }
}